// CropSplitGT_85212151153076
// MI455X (gfx1250) — compile-verified
//
#include <hip/hip_runtime.h>

// CropSplitGT forward for MI455X (gfx1250, wave32).
//
// out[h,w,n] = data[h,w,n] * (x1[n] <= w <= x2[n] && y1[n] <= h <= y2[n])
// data (512,512,400) f32, rois (4,400) f32 = [x1; y1; x2; y2].
//
// Memory-bound streaming op: 419 MB in + 419 MB out => ~36 us at 23.3 TB/s.
// No contraction dimension, so WMMA has no role. CDNA5-specific paths used:
//   * TDM tensor_load_to_lds stages the 6.4 KB ROI table into LDS in one DMA
//     (D# built per ISA ch.8; s_wait_tensorcnt before the barrier)
//   * NT-hint b128 loads/stores keep both once-touched 419 MB streams from
//     thrashing the 192 MB L2
//   * global_prefetch_b8 one grid-stride ahead (speculative => OOB-safe)

typedef __attribute__((ext_vector_type(4))) float    v4f;
typedef __attribute__((ext_vector_type(4))) unsigned u32x4;
typedef __attribute__((ext_vector_type(8))) unsigned u32x8;

#define IMG_H 512
#define IMG_W 512
#define NROI  400
#define V4_PER_PIX (NROI / 4)                 // 100 float4 per pixel
#define NV4 (IMG_H * IMG_W * V4_PER_PIX)      // 26,214,400 vec4 elements
#define THREADS 256
#define BLOCKS  5120
#define ITERS (NV4 / (THREADS * BLOCKS))      // exactly 20, no remainder

__global__ __launch_bounds__(THREADS) void crop_split_gt_kernel(
    const float* __restrict__ data,
    const float* __restrict__ rois,
    float* __restrict__ out)
{
    __shared__ float sraw[4 * NROI];   // SoA ROI table, TDM destination (6.4 KB)
    __shared__ v4f   sroi[NROI];       // AoS {x1,y1,x2,y2} for b128 LDS reads

    // ---- Wave 0: one TDM DMA moves the whole (4,400) f32 table to LDS ----
    if (threadIdx.x < 32) {            // wave32 => uniform branch for wave 0
        unsigned long long ga = (unsigned long long)rois;
        // Generic pointers to LDS keep the LDS byte offset in addr[31:0].
        unsigned lds_base = (unsigned)(unsigned long long)(void*)sraw;

        // D# group 0: count=1 (valid user descriptor), lds_addr, global_addr,
        // type=2 ("image") in bits [127:126].
        u32x4 g0;
        g0[0] = 1u;
        g0[1] = lds_base;
        g0[2] = (unsigned)ga;
        g0[3] = (unsigned)((ga >> 32) & 0x1FFFFFFull) | (2u << 30);

        // D# group 1: no multicast, data_size=2 (4 B), no pad/iterate/barrier.
        // tensor_dim0 = tile_dim0 = 1600 elems, tensor_dim1 = tile_dim1 = 1,
        // tensor_dim0_stride = 1600.
        u32x8 g1;
        g1[0] = 2u << 16;                       // data_size = 4 bytes
        g1[1] = (unsigned)(4 * NROI) << 16;     // tensor_dim0[15:0]
        g1[2] = 1u << 16;                       // tensor_dim0[31:16]=0, tensor_dim1=1
        g1[3] = (unsigned)(4 * NROI) << 16;     // tile_dim0 = 1600
        g1[4] = 1u;                             // tile_dim1 = 1, tile_dim2 = 0
        g1[5] = (unsigned)(4 * NROI);           // tensor_dim0_stride = 1600
        g1[6] = 0u;
        g1[7] = 0u;

        // Groups 2/3: zeros => tile_dim3/tile_dim4 = 0 (unused), legal 4-group form.
        u32x4 gz = (u32x4){0u, 0u, 0u, 0u};

        asm volatile("tensor_load_to_lds %0, %1, %2, %3"
                     :: "s"(g0), "s"(g1), "s"(gz), "s"(gz)
                     : "memory");
        __builtin_amdgcn_s_wait_tensorcnt(0);   // TENSORcnt==0 before barrier
    }
    __syncthreads();

    // Repack SoA -> AoS once per block so the hot loop is 4x ds_load_b128.
    for (int i = threadIdx.x; i < NROI; i += THREADS) {
        v4f r;
        r.x = sraw[0 * NROI + i];   // x1
        r.y = sraw[1 * NROI + i];   // y1
        r.z = sraw[2 * NROI + i];   // x2
        r.w = sraw[3 * NROI + i];   // y2
        sroi[i] = r;
    }
    __syncthreads();

    const v4f* __restrict__ in4  = (const v4f*)data;
    v4f* __restrict__       out4 = (v4f*)out;

    const int stride = BLOCKS * THREADS;                    // 1,310,720
    const int base   = (int)blockIdx.x * THREADS + (int)threadIdx.x;

    // Exact trip count (20): no bounds test, clean 4x unroll => 4 independent
    // NT b128 loads in flight per thread for memory-level parallelism.
#pragma unroll 4
    for (int k = 0; k < ITERS; ++k) {
        int idx = base + k * stride;

        __builtin_prefetch(&in4[idx + stride], 0, 0);       // global_prefetch_b8

        int p  = idx / V4_PER_PIX;             // pixel = h*512 + w (magic mul)
        int n0 = (idx - p * V4_PER_PIX) * 4;   // first ROI of this vec4
        float hf = (float)(p >> 9);
        float wf = (float)(p & 511);

        v4f v = __builtin_nontemporal_load(&in4[idx]);

        v4f r0 = sroi[n0 + 0];
        v4f r1 = sroi[n0 + 1];
        v4f r2 = sroi[n0 + 2];
        v4f r3 = sroi[n0 + 3];

        v4f o;
        o.x = (wf >= r0.x && wf <= r0.z && hf >= r0.y && hf <= r0.w) ? v.x : 0.0f;
        o.y = (wf >= r1.x && wf <= r1.z && hf >= r1.y && hf <= r1.w) ? v.y : 0.0f;
        o.z = (wf >= r2.x && wf <= r2.z && hf >= r2.y && hf <= r2.w) ? v.z : 0.0f;
        o.w = (wf >= r3.x && wf <= r3.z && hf >= r3.y && hf <= r3.w) ? v.w : 0.0f;

        __builtin_nontemporal_store(o, &out4[idx]);
    }
}

extern "C" void kernel_launch(void* const* d_in, const int* in_sizes, int n_in,
                              void* d_out, int out_size, void* d_ws, size_t ws_size,
                              hipStream_t stream) {
    const float* data = (const float*)d_in[0];   // (512, 512, 400) f32
    const float* rois = (const float*)d_in[1];   // (4, 400) f32
    // d_in[2] is the scalar `c`; it does not affect the GT-forward output.
    float* out = (float*)d_out;                  // (512, 512, 400) f32

    dim3 grid(BLOCKS), block(THREADS);
    hipLaunchKernelGGL(crop_split_gt_kernel, grid, block, 0, stream,
                       data, rois, out);
}